// SGCNConv_30958124270115
// MI455X (gfx1250) — compile-verified
//
#include <hip/hip_runtime.h>
#include <hip/hip_bf16.h>

#define N_NODES 50000
#define N_EDGES 600000
#define DIM 128
#define NODES_ELEMS (N_NODES * DIM)

typedef __attribute__((ext_vector_type(2))) float v2f;
typedef __attribute__((ext_vector_type(8))) float v8f;

// ---------------------------------------------------------------------------
// C[M x 128] = A[M x 128] @ B[128 x 128] (+ bias[col] if bias != nullptr)
// One wave32 per 16x16 output tile via V_WMMA_F32_16X16X4_F32.
// Block = 256 threads = 8 waves -> one 16-row x 128-col strip per block.
// gridDim.x = M/16 (M = 50000 -> 3125, exact; EXEC all-ones for WMMA).
// ---------------------------------------------------------------------------
__global__ __launch_bounds__(256) void sgcn_gemm_wmma_f32(
    const float* __restrict__ A, const float* __restrict__ B,
    const float* __restrict__ bias, float* __restrict__ C) {
  const int lane  = threadIdx.x & 31;
  const int wave  = threadIdx.x >> 5;
  const int mbase = blockIdx.x << 4;   // 16 rows per block
  const int nbase = wave << 4;         // 16 cols per wave (8 waves * 16 = 128)
  const int r15   = lane & 15;         // A: row-in-tile; B: col-in-tile
  const int khalf = lane >> 4;         // which K pair (K0/1 vs K2/3)

  v8f acc = {};
  const float* Arow = A + (size_t)(mbase + r15) * DIM + (khalf << 1);
  const float* Bcol = B + nbase + r15 + (size_t)(khalf << 1) * DIM;

  #pragma unroll 4
  for (int kk = 0; kk < DIM; kk += 4) {
    // A fragment: 16x4 f32 -> lane (m, khalf) holds {K=2*khalf, K=2*khalf+1}
    v2f a = *(const v2f*)(Arow + kk);
    // B fragment: 4x16 f32 -> lane (n, khalf) holds rows K=2*khalf, K=2*khalf+1
    v2f b;
    b.x = Bcol[(size_t)kk * DIM];
    b.y = Bcol[(size_t)kk * DIM + DIM];
    acc = __builtin_amdgcn_wmma_f32_16x16x4_f32(
        /*neg_a=*/false, a, /*neg_b=*/false, b,
        /*c_mod=*/(short)0, acc, /*reuse_a=*/false, /*reuse_b=*/false);
  }

  const float bv = bias ? bias[nbase + r15] : 0.0f;
  // C/D layout: VGPR r -> row mbase + r + 8*khalf, col nbase + (lane&15)
  float* Cp = C + (size_t)(mbase + (khalf << 3)) * DIM + nbase + r15;
  #pragma unroll
  for (int r = 0; r < 8; ++r) {
    Cp[(size_t)r * DIM] = acc[r] + bv;
  }
}

__device__ __forceinline__ float wave_reduce_add(float v) {
  #pragma unroll
  for (int off = 16; off > 0; off >>= 1) v += __shfl_xor(v, off, 32);
  return v;
}

__device__ __forceinline__ float sigmoidf(float x) {
  return 1.0f / (1.0f + __expf(-x));
}

// ---------------------------------------------------------------------------
// Self-loop gate: out = sigmoid(xl . Wg + bg) * xl      (initializes d_out)
// One wave32 per node, float4 per lane.
// ---------------------------------------------------------------------------
__global__ __launch_bounds__(256) void sgcn_loop_gate(
    const float* __restrict__ xl, const float* __restrict__ Wg,
    const float* __restrict__ bg, float* __restrict__ out) {
  const int lane = threadIdx.x & 31;
  const int node = (blockIdx.x * blockDim.x + threadIdx.x) >> 5;
  if (node >= N_NODES) return;
  const size_t base = (size_t)node * DIM + (lane << 2);
  float4 v = *(const float4*)(xl + base);
  float4 w = *(const float4*)(Wg + (lane << 2));
  float d = v.x * w.x + v.y * w.y + v.z * w.z + v.w * w.w;
  d = wave_reduce_add(d);
  const float g = sigmoidf(d + bg[0]);
  float4 o = {g * v.x, g * v.y, g * v.z, g * v.w};
  *(float4*)(out + base) = o;
}

// ---------------------------------------------------------------------------
// Edge stage: m = h[src] + b_lab[lab]; g = sigmoid(m . Wg + bg_lab[lab]);
//             atomicAdd(out[dst], g * m). One wave32 per edge.
// ---------------------------------------------------------------------------
__global__ __launch_bounds__(256) void sgcn_edge_scatter(
    const float* __restrict__ h, const int* __restrict__ src,
    const int* __restrict__ dst, const int* __restrict__ lab,
    const float* __restrict__ b_lab,   // [5][128]
    const float* __restrict__ Wg,      // [128]
    const float* __restrict__ bg_lab,  // [5]
    float* __restrict__ out) {
  const int lane = threadIdx.x & 31;
  const int e = (blockIdx.x * blockDim.x + threadIdx.x) >> 5;
  if (e >= N_EDGES) return;
  const int s = src[e];
  const int d = dst[e];
  const int l = lab[e];

  const int co = lane << 2;
  float4 hv = *(const float4*)(h + (size_t)s * DIM + co);
  float4 bv = *(const float4*)(b_lab + (size_t)l * DIM + co);
  float4 m = {hv.x + bv.x, hv.y + bv.y, hv.z + bv.z, hv.w + bv.w};
  float4 w = *(const float4*)(Wg + co);

  float dp = m.x * w.x + m.y * w.y + m.z * w.z + m.w * w.w;
  dp = wave_reduce_add(dp);
  const float g = sigmoidf(dp + bg_lab[l]);

  float* o = out + (size_t)d * DIM + co;
  atomicAdd(o + 0, g * m.x);
  atomicAdd(o + 1, g * m.y);
  atomicAdd(o + 2, g * m.z);
  atomicAdd(o + 3, g * m.w);
}

__global__ __launch_bounds__(256) void sgcn_relu(float* __restrict__ out) {
  const int i = blockIdx.x * blockDim.x + threadIdx.x;
  if (i < NODES_ELEMS) out[i] = fmaxf(out[i], 0.0f);
}

extern "C" void kernel_launch(void* const* d_in, const int* in_sizes, int n_in,
                              void* d_out, int out_size, void* d_ws, size_t ws_size,
                              hipStream_t stream) {
  const float* x        = (const float*)d_in[0];
  const int*   ei       = (const int*)d_in[1];   // [2, E] flat
  const int*   elab     = (const int*)d_in[2];
  const float* W_loop   = (const float*)d_in[3];
  const float* b_loop   = (const float*)d_in[4];
  const float* Wg_loop  = (const float*)d_in[5];
  const float* bg_loop  = (const float*)d_in[6];
  const float* W_in     = (const float*)d_in[7];
  const float* b_lab_in = (const float*)d_in[8];
  const float* Wg_in    = (const float*)d_in[9];
  const float* bg_lab_in  = (const float*)d_in[10];
  const float* W_out    = (const float*)d_in[11];
  const float* b_lab_out  = (const float*)d_in[12];
  const float* Wg_out   = (const float*)d_in[13];
  const float* bg_lab_out = (const float*)d_in[14];

  const int* e_src = ei;            // edge_index[0]
  const int* e_dst = ei + N_EDGES;  // edge_index[1]

  float* ws0 = (float*)d_ws;              // xl_raw, later h_out
  float* ws1 = ws0 + NODES_ELEMS;         // t_in, later t_out
  float* ws2 = ws1 + NODES_ELEMS;         // h_in
  float* out = (float*)d_out;

  dim3 blk(256);
  dim3 gemm_grid(N_NODES / 16);              // 3125, exact
  dim3 wave_per_node_grid((N_NODES + 7) / 8);
  dim3 wave_per_edge_grid((N_EDGES + 7) / 8);
  dim3 elem_grid((NODES_ELEMS + 255) / 256);

  // 1) xl_raw = x @ W_loop + b_loop
  sgcn_gemm_wmma_f32<<<gemm_grid, blk, 0, stream>>>(x, W_loop, b_loop, ws0);
  // 2) d_out = sigmoid(xl_raw . Wg_loop + bg_loop) * xl_raw   (init d_out)
  sgcn_loop_gate<<<wave_per_node_grid, blk, 0, stream>>>(ws0, Wg_loop, bg_loop, out);
  // 3) h_in = (x @ W_in) @ W_in
  sgcn_gemm_wmma_f32<<<gemm_grid, blk, 0, stream>>>(x, W_in, nullptr, ws1);
  sgcn_gemm_wmma_f32<<<gemm_grid, blk, 0, stream>>>(ws1, W_in, nullptr, ws2);
  // 4) h_out = (x @ W_out) @ W_out   (reuse ws1, ws0 — prior consumers done)
  sgcn_gemm_wmma_f32<<<gemm_grid, blk, 0, stream>>>(x, W_out, nullptr, ws1);
  sgcn_gemm_wmma_f32<<<gemm_grid, blk, 0, stream>>>(ws1, W_out, nullptr, ws0);
  // 5) conv_in: aggregate messages from src at dst
  sgcn_edge_scatter<<<wave_per_edge_grid, blk, 0, stream>>>(
      ws2, e_src, e_dst, elab, b_lab_in, Wg_in, bg_lab_in, out);
  // 6) conv_out: flipped edges (src/dst swapped)
  sgcn_edge_scatter<<<wave_per_edge_grid, blk, 0, stream>>>(
      ws0, e_dst, e_src, elab, b_lab_out, Wg_out, bg_lab_out, out);
  // 7) ReLU
  sgcn_relu<<<elem_grid, blk, 0, stream>>>(out);
}